// TurboQuantRotation_20813411516992
// MI455X (gfx1250) — compile-verified
//
#include <hip/hip_runtime.h>
#include <hip/hip_bf16.h>

// FWHT-4096 via H16 (x3 Kronecker factors) on CDNA5 WMMA f32 16x16x4.
// Memory-bound: one streaming pass, all transform work in LDS + WMMA.

typedef __attribute__((ext_vector_type(2))) float v2f;
typedef __attribute__((ext_vector_type(8))) float v8f;

// amdgcn builtins only exist in the device compilation pass; don't probe on host.
#if defined(__HIP_DEVICE_COMPILE__)
#if !__has_builtin(__builtin_amdgcn_wmma_f32_16x16x4_f32)
#error "device pass: missing __builtin_amdgcn_wmma_f32_16x16x4_f32"
#endif
#endif

#define FWHT_D 4096
#define IS 328   // LDS stride between i-blocks (floats); mult of 4 -> 16B alignment
#define JS 20    // LDS stride between j-rows (floats); mult of 4, bank-conflict free

__device__ __forceinline__ float hsign(int a, int b) {
    // H16[a][b] = (-1)^popcount(a&b), built branchlessly as +/-1.0f
    unsigned par = (unsigned)(__builtin_popcount(a & b) & 1);
    return __uint_as_float(0x3f800000u | (par << 31));
}

__launch_bounds__(128)
__global__ void fwht4096_wmma_kernel(const float* __restrict__ x,
                                     const float* __restrict__ signs,
                                     float* __restrict__ y)
{
    __shared__ __align__(16) float bufA[16 * IS];
    __shared__ __align__(16) float bufB[16 * IS];

    const int row    = blockIdx.x;
    const int t      = threadIdx.x;   // 0..127
    const int lane   = t & 31;
    const int wave   = t >> 5;        // 0..3
    const int half   = lane >> 4;     // 0 or 1
    const int lane16 = lane & 15;

    const float* __restrict__ xrow = x + (size_t)row * FWHT_D;
    float* __restrict__ yrow       = y + (size_t)row * FWHT_D;

    // ---- load row, apply sign diagonal, scatter into padded LDS ----
#pragma unroll
    for (int c = 0; c < 8; ++c) {
        const int p = c * 512 + t * 4;           // element index in row, mult of 4
        float4 v = *(const float4*)(xrow + p);
        float4 s = *(const float4*)(signs + p);
        v.x *= s.x; v.y *= s.y; v.z *= s.z; v.w *= s.w;
        const int i = p >> 8, j = (p >> 4) & 15, k = p & 15;   // k in {0,4,8,12}
        *(float4*)&bufA[i * IS + j * JS + k] = v;
    }
    __syncthreads();

    // ---- stage 1 (k-axis) fused with stage 2 (j-axis); tiles are wave-private ----
#pragma unroll
    for (int q = 0; q < 4; ++q) {
        const int i = wave + 4 * q;

        // stage 1: D = T_i * H   (A = T_i, B = H)
        v8f acc = {};
#pragma unroll
        for (int c = 0; c < 4; ++c) {
            const int k0 = 4 * c + 2 * half;     // K index held by vgpr p is k0+p
            v2f a, b;
            a.x = bufA[i * IS + lane16 * JS + (k0 + 0)];
            a.y = bufA[i * IS + lane16 * JS + (k0 + 1)];
            b.x = hsign(k0 + 0, lane16);
            b.y = hsign(k0 + 1, lane16);
            acc = __builtin_amdgcn_wmma_f32_16x16x4_f32(
                false, a, false, b, (short)0, acc, false, false);
        }
#pragma unroll
        for (int v = 0; v < 8; ++v) {
            const int r = v + 8 * half;          // D-matrix row for this vgpr
            bufB[i * IS + r * JS + lane16] = acc[v];
        }

        // stage 2: D = H * T1_i  (A = H, B = T1_i) — same-wave LDS, in-order
        v8f acc2 = {};
#pragma unroll
        for (int c = 0; c < 4; ++c) {
            const int k0 = 4 * c + 2 * half;
            v2f a, b;
            a.x = hsign(lane16, k0 + 0);
            a.y = hsign(lane16, k0 + 1);
            b.x = bufB[i * IS + (k0 + 0) * JS + lane16];
            b.y = bufB[i * IS + (k0 + 1) * JS + lane16];
            acc2 = __builtin_amdgcn_wmma_f32_16x16x4_f32(
                false, a, false, b, (short)0, acc2, false, false);
        }
#pragma unroll
        for (int v = 0; v < 8; ++v) {
            const int r = v + 8 * half;
            bufA[i * IS + r * JS + lane16] = acc2[v];   // T2 overwrites consumed tile
        }
    }
    __syncthreads();

    // ---- stage 3 (i-axis): mixes across tiles; D = H * U, U[i'][(j=t,k=n)] ----
#pragma unroll
    for (int q = 0; q < 4; ++q) {
        const int tt = wave + 4 * q;             // output (j) column-block
        v8f acc3 = {};
#pragma unroll
        for (int c = 0; c < 4; ++c) {
            const int k0 = 4 * c + 2 * half;     // i' indices k0+p
            v2f a, b;
            a.x = hsign(lane16, k0 + 0);
            a.y = hsign(lane16, k0 + 1);
            b.x = bufA[(k0 + 0) * IS + tt * JS + lane16];
            b.y = bufA[(k0 + 1) * IS + tt * JS + lane16];
            acc3 = __builtin_amdgcn_wmma_f32_16x16x4_f32(
                false, a, false, b, (short)0, acc3, false, false);
        }
#pragma unroll
        for (int v = 0; v < 8; ++v) {
            const int i = v + 8 * half;          // output i index
            bufB[i * IS + tt * JS + lane16] = acc3[v];
        }
    }
    __syncthreads();

    // ---- coalesced store ----
#pragma unroll
    for (int c = 0; c < 8; ++c) {
        const int p = c * 512 + t * 4;
        const int i = p >> 8, j = (p >> 4) & 15, k = p & 15;
        float4 v = *(const float4*)&bufB[i * IS + j * JS + k];
        *(float4*)(yrow + p) = v;
    }
}

extern "C" void kernel_launch(void* const* d_in, const int* in_sizes, int n_in,
                              void* d_out, int out_size, void* d_ws, size_t ws_size,
                              hipStream_t stream) {
    const float* x     = (const float*)d_in[0];
    const float* signs = (const float*)d_in[1];
    float* out         = (float*)d_out;

    const int total = in_sizes[0];          // 4*4096*4096
    const int rows  = total / FWHT_D;       // 16384
    fwht4096_wmma_kernel<<<rows, 128, 0, stream>>>(x, signs, out);
}